// GCN_model_79413945303589
// MI455X (gfx1250) — compile-verified
//
#include <hip/hip_runtime.h>

typedef __attribute__((ext_vector_type(2))) float v2f;
typedef __attribute__((ext_vector_type(8))) float v8f;

#define HIDDEN 6
#define NCLS   10
#define NGRAPH 64
#define LDH    16   // padded feature stride (64B rows -> aligned float4)

__device__ __forceinline__ int imin(int a, int b) { return a < b ? a : b; }

// ---------------------------------------------------------------- utilities

__global__ void fill_kernel(float* __restrict__ p, float v, int n) {
    int i = blockIdx.x * blockDim.x + threadIdx.x;
    if (i < n) p[i] = v;
}

// deg[dst] += 1 per edge (deg pre-initialized to 1.0 for the self loop)
__global__ void deg_kernel(const long long* __restrict__ dst,
                           float* __restrict__ deg, int E) {
    int e = blockIdx.x * blockDim.x + threadIdx.x;
    if (e < E) unsafeAtomicAdd(&deg[(int)dst[e]], 1.0f);
}

__global__ void rsqrt_kernel(float* __restrict__ d, int N) {
    int i = blockIdx.x * blockDim.x + threadIdx.x;
    if (i < N) d[i] = rsqrtf(d[i]);
}

__global__ void norm_kernel(const long long* __restrict__ src,
                            const long long* __restrict__ dst,
                            const float* __restrict__ dinv,
                            float* __restrict__ nrm, int E) {
    int e = blockIdx.x * blockDim.x + threadIdx.x;
    if (e < E) nrm[e] = dinv[(int)src[e]] * dinv[(int)dst[e]];
}

// -------------------------- W transpose + zero-pad:  Wt[n][k], 16 x KPAD
// Wt[n*KPAD + k] = (n < FOUT && k < FIN) ? W[k*FOUT + n] : 0
template<int FIN, int FOUT>
__global__ void wtrans_kernel(const float* __restrict__ W,
                              float* __restrict__ Wt) {
    constexpr int KPAD = (FIN + 3) & ~3;
    int idx = blockIdx.x * blockDim.x + threadIdx.x;
    if (idx >= 16 * KPAD) return;
    int n = idx / KPAD;
    int k = idx % KPAD;
    Wt[idx] = (n < FOUT && k < FIN) ? W[k * FOUT + n] : 0.f;
}

// ------------------------------------------------- WMMA tile GEMM (f32 K=4)
// T[m, :] = act(X[m, :]) @ W,  T stored with stride LDH.  One wave / 16 rows.
// V_WMMA_F32_16X16X4_F32 fragments:
//   A 16x4:  lanes 0-15: M=lane, VGPR{0,1}=K{0,1}; lanes 16-31: K{2,3}
//   B 4x16:  lanes 0-15: N=lane, VGPR{0,1}=K{0,1}; lanes 16-31: K{2,3}
//   C/D:     VGPR r = row r (lanes 0-15) / row 8+r (lanes 16-31)
// No masks in the hot loop:
//   - B comes from the zero-padded transposed Wt (contiguous pair per lane)
//   - A pad cols (small layers) are genuine zeros in the stride-16 buffers
//   - duplicate (clamped) A rows only affect D rows >= N, never stored
template<int FIN, int LDX, bool RELU>
__global__ void wmma_gemm_kernel(const float* __restrict__ X,
                                 const float* __restrict__ Wt,
                                 float* __restrict__ T, int N) {
    constexpr int KPAD = (FIN + 3) & ~3;
    const int row0 = blockIdx.x * 16;
    const int lane = threadIdx.x;
    const int l    = lane & 15;
    const int hi   = lane >> 4;

    const int m  = row0 + l;
    const int mc = imin(m, N - 1);
    const float* __restrict__ xrow = X + (long)mc * LDX;
    const float* __restrict__ wrow = Wt + l * KPAD;

    v8f acc = {};
#pragma unroll
    for (int k0 = 0; k0 < KPAD; k0 += 4) {
        const int ka = k0 + hi * 2;
        float a0 = xrow[ka];
        float a1 = xrow[ka + 1];
        if constexpr (RELU) { a0 = fmaxf(a0, 0.f); a1 = fmaxf(a1, 0.f); }
        float b0 = wrow[ka];
        float b1 = wrow[ka + 1];
        v2f a; a.x = a0; a.y = a1;
        v2f b; b.x = b0; b.y = b1;
        acc = __builtin_amdgcn_wmma_f32_16x16x4_f32(
            false, a, false, b, (short)0, acc, false, false);
    }

#pragma unroll
    for (int r = 0; r < 8; ++r) {
        int mr = row0 + hi * 8 + r;
        if (mr < N) T[(long)mr * LDH + l] = acc[r];
    }
}

// ------------------------------- self-loop + bias init: O = b + dinv^2 * T
// One thread per (node, float4 quad); pad cols are written as exact zeros.
template<int FOUT>
__global__ void selfloop_init_kernel(const float* __restrict__ T,
                                     const float* __restrict__ dinv,
                                     const float* __restrict__ bias,
                                     float* __restrict__ O, int N) {
    int t = blockIdx.x * blockDim.x + threadIdx.x;
    int i = t >> 2;
    int q = t & 3;
    if (i >= N) return;
    float w = dinv[i] * dinv[i];
    float4 v = ((const float4*)(T + (long)i * LDH))[q];
    int f0 = q * 4;
    float4 o;
    o.x = (f0 + 0 < FOUT) ? bias[imin(f0 + 0, FOUT - 1)] + w * v.x : 0.f;
    o.y = (f0 + 1 < FOUT) ? bias[imin(f0 + 1, FOUT - 1)] + w * v.y : 0.f;
    o.z = (f0 + 2 < FOUT) ? bias[imin(f0 + 2, FOUT - 1)] + w * v.z : 0.f;
    o.w = (f0 + 3 < FOUT) ? bias[imin(f0 + 3, FOUT - 1)] + w * v.w : 0.f;
    ((float4*)(O + (long)i * LDH))[q] = o;
}

// ------------------------------------ edge scatter: O[dst] += norm * T[src]
template<int FOUT>
__global__ void edge_agg_kernel(const long long* __restrict__ src,
                                const long long* __restrict__ dst,
                                const float* __restrict__ nrm,
                                const float* __restrict__ T,
                                float* __restrict__ O, int E) {
    int e = blockIdx.x * blockDim.x + threadIdx.x;
    if (e >= E) return;
    int s = (int)src[e];
    int d = (int)dst[e];
    float w = nrm[e];
    const float4* __restrict__ ts = (const float4*)(T + (long)s * LDH);
    float* __restrict__ od = O + (long)d * LDH;

    float4 v0 = ts[0];
    unsafeAtomicAdd(&od[0], w * v0.x);
    unsafeAtomicAdd(&od[1], w * v0.y);
    unsafeAtomicAdd(&od[2], w * v0.z);
    unsafeAtomicAdd(&od[3], w * v0.w);
    float4 v1 = ts[1];
    unsafeAtomicAdd(&od[4], w * v1.x);
    unsafeAtomicAdd(&od[5], w * v1.y);
    if constexpr (FOUT > 6) {
        unsafeAtomicAdd(&od[6], w * v1.z);
        unsafeAtomicAdd(&od[7], w * v1.w);
        float4 v2 = ts[2];
        unsafeAtomicAdd(&od[8], w * v2.x);
        unsafeAtomicAdd(&od[9], w * v2.y);
    }
}

// --------------------------------------------- graph pooling (sum + count)
__global__ void pool_kernel(const float* __restrict__ H,
                            const long long* __restrict__ batch,
                            float* __restrict__ pool, float* __restrict__ cnt,
                            int N) {
    int i = blockIdx.x * blockDim.x + threadIdx.x;
    if (i >= N) return;
    int g = (int)batch[i];
    const float4* __restrict__ h4 = (const float4*)(H + (long)i * LDH);
    float4 a = h4[0], b = h4[1], c = h4[2];
    float* pg = pool + g * LDH;
    unsafeAtomicAdd(&pg[0], fmaxf(a.x, 0.f));
    unsafeAtomicAdd(&pg[1], fmaxf(a.y, 0.f));
    unsafeAtomicAdd(&pg[2], fmaxf(a.z, 0.f));
    unsafeAtomicAdd(&pg[3], fmaxf(a.w, 0.f));
    unsafeAtomicAdd(&pg[4], fmaxf(b.x, 0.f));
    unsafeAtomicAdd(&pg[5], fmaxf(b.y, 0.f));
    unsafeAtomicAdd(&pg[6], fmaxf(b.z, 0.f));
    unsafeAtomicAdd(&pg[7], fmaxf(b.w, 0.f));
    unsafeAtomicAdd(&pg[8], fmaxf(c.x, 0.f));
    unsafeAtomicAdd(&pg[9], fmaxf(c.y, 0.f));
    unsafeAtomicAdd(&cnt[g], 1.0f);
}

// ----------------------------------------------- mean + log_softmax, 64 thr
__global__ void lsm_kernel(const float* __restrict__ pool,
                           const float* __restrict__ cnt,
                           float* __restrict__ out) {
    int g = threadIdx.x;
    if (g < NGRAPH) {
        float c = fmaxf(cnt[g], 1.0f);
        float p[NCLS];
        float m = -3.0e38f;
        for (int j = 0; j < NCLS; ++j) {
            p[j] = pool[g * LDH + j] / c;
            m = fmaxf(m, p[j]);
        }
        float s = 0.f;
        for (int j = 0; j < NCLS; ++j) s += __expf(p[j] - m);
        float lse = m + __logf(s);
        for (int j = 0; j < NCLS; ++j) out[g * NCLS + j] = p[j] - lse;
    }
}

// ================================================================= launcher

extern "C" void kernel_launch(void* const* d_in, const int* in_sizes, int n_in,
                              void* d_out, int out_size, void* d_ws, size_t ws_size,
                              hipStream_t stream) {
    const float*     x     = (const float*)d_in[0];
    const long long* ei    = (const long long*)d_in[1];
    const long long* batch = (const long long*)d_in[2];
    const float* W1 = (const float*)d_in[3];  const float* b1 = (const float*)d_in[4];
    const float* W2 = (const float*)d_in[5];  const float* b2 = (const float*)d_in[6];
    const float* W3 = (const float*)d_in[7];  const float* b3 = (const float*)d_in[8];
    const float* W4 = (const float*)d_in[9];  const float* b4 = (const float*)d_in[10];
    const float* Wf = (const float*)d_in[11]; const float* bf = (const float*)d_in[12];

    const int F = 128;
    const int N = in_sizes[0] / F;
    const int E = in_sizes[1] / 2;

    const long long* src = ei;
    const long long* dst = ei + E;

    // workspace layout (floats)
    float* ws   = (float*)d_ws;
    float* dinv = ws;                       // N      (holds deg, then rsqrt)
    float* nrm  = dinv + N;                 // E
    float* bufT = nrm + E;                  // N*LDH
    float* bufA = bufT + (long)N * LDH;     // N*LDH
    float* bufB = bufA + (long)N * LDH;     // N*LDH
    float* pool = bufB + (long)N * LDH;     // NGRAPH*LDH
    float* cnt  = pool + NGRAPH * LDH;      // NGRAPH
    float* wt   = cnt + NGRAPH;             // 16*128 scratch (reused per layer)

    const int TB = 256;
    dim3 blk(TB);
    auto g1  = [&](long n) { return dim3((unsigned)((n + TB - 1) / TB)); };
    dim3 gE  = g1(E);
    dim3 gN  = g1(N);
    dim3 gN4 = g1((long)N * 4);
    dim3 gT((N + 15) / 16);
    dim3 w32(32);
    dim3 gW1 = g1(16 * 128);
    dim3 gWs = g1(16 * 8);

    // ---- normalization: deg (self loop => init 1), dinv, edge norms
    fill_kernel<<<gN, blk, 0, stream>>>(dinv, 1.0f, N);
    deg_kernel<<<gE, blk, 0, stream>>>(dst, dinv, E);
    rsqrt_kernel<<<gN, blk, 0, stream>>>(dinv, N);
    norm_kernel<<<gE, blk, 0, stream>>>(src, dst, dinv, nrm, E);

    // ---- layer 1: x(128) -> bufA(6)
    wtrans_kernel<128, HIDDEN><<<gW1, blk, 0, stream>>>(W1, wt);
    wmma_gemm_kernel<128, 128, false><<<gT, w32, 0, stream>>>(x, wt, bufT, N);
    selfloop_init_kernel<HIDDEN><<<gN4, blk, 0, stream>>>(bufT, dinv, b1, bufA, N);
    edge_agg_kernel<HIDDEN><<<gE, blk, 0, stream>>>(src, dst, nrm, bufT, bufA, E);

    // ---- layer 2: bufA -> bufB
    wtrans_kernel<HIDDEN, HIDDEN><<<gWs, blk, 0, stream>>>(W2, wt);
    wmma_gemm_kernel<HIDDEN, LDH, true><<<gT, w32, 0, stream>>>(bufA, wt, bufT, N);
    selfloop_init_kernel<HIDDEN><<<gN4, blk, 0, stream>>>(bufT, dinv, b2, bufB, N);
    edge_agg_kernel<HIDDEN><<<gE, blk, 0, stream>>>(src, dst, nrm, bufT, bufB, E);

    // ---- layer 3: bufB -> bufA
    wtrans_kernel<HIDDEN, HIDDEN><<<gWs, blk, 0, stream>>>(W3, wt);
    wmma_gemm_kernel<HIDDEN, LDH, true><<<gT, w32, 0, stream>>>(bufB, wt, bufT, N);
    selfloop_init_kernel<HIDDEN><<<gN4, blk, 0, stream>>>(bufT, dinv, b3, bufA, N);
    edge_agg_kernel<HIDDEN><<<gE, blk, 0, stream>>>(src, dst, nrm, bufT, bufA, E);

    // ---- layer 4: bufA -> bufB
    wtrans_kernel<HIDDEN, HIDDEN><<<gWs, blk, 0, stream>>>(W4, wt);
    wmma_gemm_kernel<HIDDEN, LDH, true><<<gT, w32, 0, stream>>>(bufA, wt, bufT, N);
    selfloop_init_kernel<HIDDEN><<<gN4, blk, 0, stream>>>(bufT, dinv, b4, bufB, N);
    edge_agg_kernel<HIDDEN><<<gE, blk, 0, stream>>>(src, dst, nrm, bufT, bufB, E);

    // ---- layer 5: bufB -> bufA (fout = 10)
    wtrans_kernel<HIDDEN, NCLS><<<gWs, blk, 0, stream>>>(Wf, wt);
    wmma_gemm_kernel<HIDDEN, LDH, true><<<gT, w32, 0, stream>>>(bufB, wt, bufT, N);
    selfloop_init_kernel<NCLS><<<gN4, blk, 0, stream>>>(bufT, dinv, bf, bufA, N);
    edge_agg_kernel<NCLS><<<gE, blk, 0, stream>>>(src, dst, nrm, bufT, bufA, E);

    // ---- pooling + log_softmax
    fill_kernel<<<g1(NGRAPH * LDH), blk, 0, stream>>>(pool, 0.0f, NGRAPH * LDH);
    fill_kernel<<<g1(NGRAPH), blk, 0, stream>>>(cnt, 0.0f, NGRAPH);
    pool_kernel<<<gN, blk, 0, stream>>>(bufA, batch, pool, cnt, N);
    lsm_kernel<<<dim3(1), dim3(64), 0, stream>>>(pool, cnt, (float*)d_out);
}